// normGNN_5016521802571
// MI455X (gfx1250) — compile-verified
//
#include <hip/hip_runtime.h>
#include <hip/hip_bf16.h>

typedef __attribute__((ext_vector_type(2))) float v2f;
typedef __attribute__((ext_vector_type(8))) float v8f;

#define DIM 128

// ---------------------------------------------------------------------------
// Y[M x 128] = act(X[M x 128] @ W^T + bias), W stored row-major [n][k].
// One wave per 16x16 tile (V_WMMA_F32_16X16X4_F32, K loop of 32 steps).
// Block = 256 threads = 8 waves = 16 rows x 128 cols.
// ---------------------------------------------------------------------------
__global__ __launch_bounds__(256) void gemm_bias_act(
    const float* __restrict__ X, const float* __restrict__ W,
    const float* __restrict__ bias, float* __restrict__ Y, int do_relu)
{
    const int wave = threadIdx.x >> 5;
    const int lane = threadIdx.x & 31;
    const int l15  = lane & 15;
    const int hi   = lane >> 4;            // 0: K={0,1}, 1: K={2,3}
    const int row0 = blockIdx.x * 16;
    const int col0 = wave * 16;

    const float* xr = X + (size_t)(row0 + l15) * DIM + 2 * hi;
    const float* wr = W + (size_t)(col0 + l15) * DIM + 2 * hi;

    v8f c = {};
#pragma unroll
    for (int k = 0; k < DIM; k += 4) {
        v2f a = *(const v2f*)(xr + k);
        v2f b = *(const v2f*)(wr + k);
        c = __builtin_amdgcn_wmma_f32_16x16x4_f32(
                false, a, false, b, (short)0, c, false, false);
    }

    const int col = col0 + l15;
    const float bv = bias[col];
#pragma unroll
    for (int r = 0; r < 8; ++r) {
        const int row = row0 + r + hi * 8;
        float v = c[r] + bv;
        if (do_relu) v = fmaxf(v, 0.0f);
        Y[(size_t)row * DIM + col] = v;
    }
}

// ---------------------------------------------------------------------------
// Edge phase: agg[dst] += h[src] * w. One wave per edge; each lane moves a
// float4 (4 x 32 = 128 features). HW fp32 atomics (global_atomic_add_f32).
// ---------------------------------------------------------------------------
__global__ __launch_bounds__(256) void gather_scale_scatter(
    const float* __restrict__ H, const int* __restrict__ src,
    const int* __restrict__ dst, const float* __restrict__ ew,
    float* __restrict__ AGG, int E)
{
    const int gid  = blockIdx.x * blockDim.x + threadIdx.x;
    const int e    = gid >> 5;
    if (e >= E) return;
    const int lane = gid & 31;

    const int   s = src[e];
    const int   d = dst[e];
    const float w = ew[e];

    const float4 v = *(const float4*)(H + (size_t)s * DIM + lane * 4);
    float* o = AGG + (size_t)d * DIM + lane * 4;
    unsafeAtomicAdd(o + 0, v.x * w);
    unsafeAtomicAdd(o + 1, v.y * w);
    unsafeAtomicAdd(o + 2, v.z * w);
    unsafeAtomicAdd(o + 3, v.w * w);
}

// ---------------------------------------------------------------------------
// Fused GraphConv layer:
//   t = agg @ Wrel^T + brel + h @ Wroot^T          (two WMMA accumulations)
//   t = LayerNorm(t) * g + b                        (block-local, LDS stage)
//   h' = relu(t [+ h2])
// Block = 16 rows x 128 cols; 8 waves each own a 16x16 tile.
// ---------------------------------------------------------------------------
__global__ __launch_bounds__(256) void graphconv_ln_act(
    const float* __restrict__ AGG, const float* __restrict__ H,
    const float* __restrict__ Wrel, const float* __restrict__ brel,
    const float* __restrict__ Wroot,
    const float* __restrict__ lng, const float* __restrict__ lnb,
    const float* __restrict__ H2, float* __restrict__ Hout, int use_res)
{
    __shared__ float tile[16][DIM + 2];   // +2 pad to dodge bank conflicts

    const int wave = threadIdx.x >> 5;
    const int lane = threadIdx.x & 31;
    const int l15  = lane & 15;
    const int hi   = lane >> 4;
    const int row0 = blockIdx.x * 16;
    const int col0 = wave * 16;

    const float* ar = AGG  + (size_t)(row0 + l15) * DIM + 2 * hi;
    const float* hr = H    + (size_t)(row0 + l15) * DIM + 2 * hi;
    const float* w1 = Wrel  + (size_t)(col0 + l15) * DIM + 2 * hi;
    const float* w2 = Wroot + (size_t)(col0 + l15) * DIM + 2 * hi;

    v8f c = {};
#pragma unroll
    for (int k = 0; k < DIM; k += 4) {           // agg @ Wrel^T
        v2f a = *(const v2f*)(ar + k);
        v2f b = *(const v2f*)(w1 + k);
        c = __builtin_amdgcn_wmma_f32_16x16x4_f32(
                false, a, false, b, (short)0, c, false, false);
    }
#pragma unroll
    for (int k = 0; k < DIM; k += 4) {           // + h @ Wroot^T
        v2f a = *(const v2f*)(hr + k);
        v2f b = *(const v2f*)(w2 + k);
        c = __builtin_amdgcn_wmma_f32_16x16x4_f32(
                false, a, false, b, (short)0, c, false, false);
    }

    const float bv = brel[col0 + l15];
#pragma unroll
    for (int r = 0; r < 8; ++r)
        tile[r + 8 * hi][col0 + l15] = c[r] + bv;

    __syncthreads();

    // LayerNorm: 16 consecutive threads own one row (stay in one half-wave).
    const int row = threadIdx.x >> 4;    // 0..15
    const int seg = threadIdx.x & 15;    // 0..15, 8 cols each

    float vals[8];
    float s = 0.0f, ss = 0.0f;
#pragma unroll
    for (int j = 0; j < 8; ++j) {
        float v = tile[row][seg * 8 + j];
        vals[j] = v;
        s  += v;
        ss += v * v;
    }
#pragma unroll
    for (int off = 1; off < 16; off <<= 1) {
        s  += __shfl_xor(s,  off, 16);
        ss += __shfl_xor(ss, off, 16);
    }
    const float mu   = s * (1.0f / DIM);
    const float var  = ss * (1.0f / DIM) - mu * mu;
    const float rstd = rsqrtf(var + 1e-5f);

    const size_t grow = (size_t)(row0 + row) * DIM;
#pragma unroll
    for (int j = 0; j < 8; ++j) {
        const int col = seg * 8 + j;
        float v = (vals[j] - mu) * rstd * lng[col] + lnb[col];
        if (use_res) v += H2[grow + col];
        v = fmaxf(v, 0.0f);
        Hout[grow + col] = v;
    }
}

// ---------------------------------------------------------------------------
extern "C" void kernel_launch(void* const* d_in, const int* in_sizes, int n_in,
                              void* d_out, int out_size, void* d_ws, size_t ws_size,
                              hipStream_t stream) {
    const float* x      = (const float*)d_in[0];
    const int*   edge   = (const int*)  d_in[1];   // [2, E]
    const float* ew     = (const float*)d_in[2];
    const float* W_in   = (const float*)d_in[3];
    const float* b_in   = (const float*)d_in[4];
    const float* W_rel  = (const float*)d_in[5];   // [3, D, D]
    const float* b_rel  = (const float*)d_in[6];   // [3, D]
    const float* W_root = (const float*)d_in[7];   // [3, D, D]
    const float* ln_g   = (const float*)d_in[8];
    const float* ln_b   = (const float*)d_in[9];
    const float* W_out  = (const float*)d_in[10];
    const float* b_out  = (const float*)d_in[11];

    const int N = 50000, E = 800000, D = DIM;
    const size_t nd = (size_t)N * D;

    float* out_h   = (float*)d_out;        // first N*D: final h
    float* out_mat = out_h + nd;           // second N*D: out

    float* h0  = (float*)d_ws;             // post input layer (== h2 for layer 1)
    float* agg = h0  + nd;
    float* hA  = agg + nd;
    float* hB  = hA  + nd;

    const int* srcv = edge;
    const int* dstv = edge + E;

    const int gemm_blocks = N / 16;             // 3125
    const int scat_blocks = (E * 32) / 256;     // 100000

    // h0 = relu(x @ W_in^T + b_in)
    gemm_bias_act<<<gemm_blocks, 256, 0, stream>>>(x, W_in, b_in, h0, 1);

    // ---- layer 0 (no residual) ----
    hipMemsetAsync(agg, 0, nd * sizeof(float), stream);
    gather_scale_scatter<<<scat_blocks, 256, 0, stream>>>(h0, srcv, dstv, ew, agg, E);
    graphconv_ln_act<<<gemm_blocks, 256, 0, stream>>>(
        agg, h0, W_rel, b_rel, W_root, ln_g, ln_b, nullptr, hA, 0);

    // ---- layer 1 (residual with h2 = h0) ----
    hipMemsetAsync(agg, 0, nd * sizeof(float), stream);
    gather_scale_scatter<<<scat_blocks, 256, 0, stream>>>(hA, srcv, dstv, ew, agg, E);
    graphconv_ln_act<<<gemm_blocks, 256, 0, stream>>>(
        agg, hA, W_rel + (size_t)D * D, b_rel + D, W_root + (size_t)D * D,
        ln_g + D, ln_b + D, h0, hB, 1);

    // ---- layer 2 (no residual) -> final h straight into d_out ----
    hipMemsetAsync(agg, 0, nd * sizeof(float), stream);
    gather_scale_scatter<<<scat_blocks, 256, 0, stream>>>(hB, srcv, dstv, ew, agg, E);
    graphconv_ln_act<<<gemm_blocks, 256, 0, stream>>>(
        agg, hB, W_rel + 2 * (size_t)D * D, b_rel + 2 * D, W_root + 2 * (size_t)D * D,
        ln_g + 2 * D, ln_b + 2 * D, nullptr, out_h, 0);

    // out = h @ W_out^T + b_out (no relu)
    gemm_bias_act<<<gemm_blocks, 256, 0, stream>>>(out_h, W_out, b_out, out_mat, 0);
}